// CrossLayer_74491912782062
// MI455X (gfx1250) — compile-verified
//
#include <hip/hip_runtime.h>

// DCN cross layer, B=262144 rows, D=256, L=3, f32 in/out.
// Roofline: 512 MB compulsory traffic @ 23.3 TB/s => ~22us floor => pure streaming
// kernel. Wave32 handles a PAIR of rows (lanes 0-15 -> row A, 16-31 -> row B),
// 16 elements per lane, b128 NT loads/stores (stream exceeds 192MB L2).
//
// Layer sums satisfy s_{i+1} = s_i*dot(W_i,x0) + sum(b_i) + sum(x0), so per row we
// need only 3 independent reductions (t, p0, p1) and one final elementwise pass:
// out = (s2*W2 + 1)*x0 + b2   (2 FMAs/element). Reductions are 4-step ds_swizzle
// xor-trees confined to each 16-lane half (immediate offsets, no index VGPRs).

typedef float v4f __attribute__((ext_vector_type(4)));

#define DIM 256
#define PAIRS_PER_WAVE 4    // 8 rows per wave
#define WAVES_PER_BLOCK 8   // 256 threads

// One butterfly step: ds_swizzle group-of-32, src = lane ^ XM (and_mask=0x1f).
template <int XM>
__device__ __forceinline__ float bf_add(float v) {
    int s = __builtin_amdgcn_ds_swizzle(__float_as_int(v), (XM << 10) | 0x1f);
    return v + __int_as_float(s);
}

// Sum across a 16-lane half (xor 1,2,4,8 never crosses lane bit 4).
__device__ __forceinline__ float group16_sum(float v) {
    v = bf_add<1>(v);
    v = bf_add<2>(v);
    v = bf_add<4>(v);
    v = bf_add<8>(v);
    return v;
}

__device__ __forceinline__ float hsum4(const v4f& a) {
    return (a.x + a.y) + (a.z + a.w);
}

__device__ __forceinline__ float dot4_acc(float r, const v4f& w, const v4f& a) {
    r = fmaf(w.x, a.x, r);
    r = fmaf(w.y, a.y, r);
    r = fmaf(w.z, a.z, r);
    r = fmaf(w.w, a.w, r);
    return r;
}

__global__ __launch_bounds__(256) void cross_layer_kernel(
    const float* __restrict__ x,
    const float* __restrict__ W,
    const float* __restrict__ b,
    float* __restrict__ out,
    int B)
{
    const int lane = threadIdx.x & 31;
    const int sub  = lane & 15;       // position within 16-lane half
    const int half = lane >> 4;       // 0 -> row A, 1 -> row B
    const long waveId = (long)blockIdx.x * WAVES_PER_BLOCK + (threadIdx.x >> 5);
    const long pairBase = waveId * PAIRS_PER_WAVE;

    // Chunk j covers row elements [64*j + sub*4, +4): 4 chunks = 16 elems/lane.
    const int eb = sub * 4;

    // Per-wave preamble: W (3KB, L2-resident, reused for 8 rows) and final bias.
    v4f w[3][4], bL[4];
#pragma unroll
    for (int i = 0; i < 3; ++i)
#pragma unroll
        for (int j = 0; j < 4; ++j)
            w[i][j] = *(const v4f*)(W + i * DIM + 64 * j + eb);
#pragma unroll
    for (int j = 0; j < 4; ++j)
        bL[j] = *(const v4f*)(b + 2 * DIM + 64 * j + eb);

    // Row-independent bias sums q0 = sum(b0), q1 = sum(b1) (16-lane trees: both
    // halves hold identical chunks, so a 32-lane sum would double-count).
    float q0p = 0.f, q1p = 0.f;
#pragma unroll
    for (int j = 0; j < 4; ++j) {
        q0p += hsum4(*(const v4f*)(b + 0 * DIM + 64 * j + eb));
        q1p += hsum4(*(const v4f*)(b + 1 * DIM + 64 * j + eb));
    }
    const float q0 = group16_sum(q0p);
    const float q1 = group16_sum(q1p);

#pragma unroll
    for (int p = 0; p < PAIRS_PER_WAVE; ++p) {
        const long rowA = (pairBase + p) * 2;
        if (rowA >= B) break;
        const long myRow = rowA + half;
        const long ldRow = (myRow < B) ? myRow : (long)(B - 1);  // clamp (no OOB read)
        const float* xr = x + ldRow * DIM;

        // Prefetch next pair (gfx1250 global_prefetch_b8; speculative, safe).
        if (p + 1 < PAIRS_PER_WAVE) {
            const long pfRow = (ldRow + 2 < B) ? ldRow + 2 : (long)(B - 1);
            const float* pf = x + pfRow * DIM;
#pragma unroll
            for (int j = 0; j < 4; ++j)
                __builtin_prefetch(pf + 64 * j + eb, 0, 0);
        }

        // Streaming (non-temporal) loads.
        v4f a[4];
#pragma unroll
        for (int j = 0; j < 4; ++j)
            a[j] = __builtin_nontemporal_load((const v4f*)(xr + 64 * j + eb));

        // Lane-local partials, then 3 independent 4-step half-wave trees.
        float tp = 0.f, p0p = 0.f, p1p = 0.f;
#pragma unroll
        for (int j = 0; j < 4; ++j) {
            tp  += hsum4(a[j]);
            p0p  = dot4_acc(p0p, w[0][j], a[j]);
            p1p  = dot4_acc(p1p, w[1][j], a[j]);
        }
        const float t  = group16_sum(tp);   // sum(x0)   for this half's row
        const float p0 = group16_sum(p0p);  // dot(W0,x0)
        const float p1 = group16_sum(p1p);  // dot(W1,x0)

        // Scalar layer-sum recurrence.
        const float s1 = fmaf(t,  p0, q0 + t);
        const float s2 = fmaf(s1, p1, q1 + t);

        // Final layer, 2 FMAs/element: r = s2*W2 + 1;  out = r*x0 + b2.
        v4f c[4];
#pragma unroll
        for (int j = 0; j < 4; ++j) {
            v4f r;
            r.x = fmaf(s2, w[2][j].x, 1.0f);
            r.y = fmaf(s2, w[2][j].y, 1.0f);
            r.z = fmaf(s2, w[2][j].z, 1.0f);
            r.w = fmaf(s2, w[2][j].w, 1.0f);
            c[j].x = fmaf(r.x, a[j].x, bL[j].x);
            c[j].y = fmaf(r.y, a[j].y, bL[j].y);
            c[j].z = fmaf(r.z, a[j].z, bL[j].z);
            c[j].w = fmaf(r.w, a[j].w, bL[j].w);
        }

        if (myRow < B) {
            float* orow = out + myRow * DIM;
#pragma unroll
            for (int j = 0; j < 4; ++j)
                __builtin_nontemporal_store(c[j], (v4f*)(orow + 64 * j + eb));
        }
    }
}

extern "C" void kernel_launch(void* const* d_in, const int* in_sizes, int n_in,
                              void* d_out, int out_size, void* d_ws, size_t ws_size,
                              hipStream_t stream) {
    const float* x = (const float*)d_in[0];  // (B, 1, 256) f32
    const float* W = (const float*)d_in[1];  // (3, 256)    f32
    const float* b = (const float*)d_in[2];  // (3, 256)    f32
    float* out = (float*)d_out;              // (B, 256)    f32

    const int B = in_sizes[0] / DIM;
    const int rowsPerBlock = WAVES_PER_BLOCK * PAIRS_PER_WAVE * 2;  // 64
    const int grid = (B + rowsPerBlock - 1) / rowsPerBlock;         // 4096

    cross_layer_kernel<<<grid, 256, 0, stream>>>(x, W, b, out, B);
}